// Radon_31507880083522
// MI455X (gfx1250) — compile-verified
//
#include <hip/hip_runtime.h>

// ---------------------------------------------------------------------------
// Radon transform for x:(2,1,512,512) f32, theta: 180 integer degrees.
// out[n, 0, w, a] = sum_h bilinear(x[n,0], rotate(theta[a]) @ grid(h, w))
// Strategy: per-(n,a) workgroup; stage 64-row bands (+1 halo row) of the image
// into LDS with the CDNA5 Tensor Data Mover (double buffered, TENSORcnt),
// each thread (one per w) walks the analytic h-interval of the line that
// intersects the current band and bilinearly gathers from LDS.
// ---------------------------------------------------------------------------

typedef unsigned int tdm_v4u __attribute__((ext_vector_type(4)));
typedef int          tdm_v8i __attribute__((ext_vector_type(8)));
typedef int          tdm_v4i __attribute__((ext_vector_type(4)));

#define RADON_H   512
#define RADON_W   512
#define RADON_A   180
#define RADON_N   2
#define BAND_R    64
#define BAND_ROWS (BAND_R + 1)              // 65 rows incl. 1-row halo
#define BUF_FLOATS (BAND_ROWS * RADON_W)    // 33280 floats
#define BUF_BYTES  (BUF_FLOATS * 4)         // 133120 bytes
#define NBANDS     (RADON_H / BAND_R)       // 8

// Issue one TDM tensor load: tile = 512 x 65 rows of f32 from global -> LDS.
// tensor_dim1 = rows remaining in the image from this band's start, so the
// halo row past the last image row is zero-filled by the TDM (OOB-read = 0),
// which implements padding_mode='zeros' for the bottom edge for free.
__device__ __forceinline__ void tdm_load_band(const float* gsrc,
                                              unsigned lds_byte_off,
                                              int rows_remaining) {
  unsigned long long ga = (unsigned long long)(size_t)gsrc;
  tdm_v4u g0;
  g0.x = 1u;                                    // count=1 (valid), user mode
  g0.y = lds_byte_off;                          // lds_addr (bytes)
  g0.z = (unsigned)ga;                          // global_addr[31:0]
  g0.w = (unsigned)((ga >> 32) & 0x1FFFFFFull)  // global_addr[56:32]
         | 0x80000000u;                         // type = 2 ("image")
  tdm_v8i g1;
  g1[0] = 0x00020000;                           // data_size=2 (4B); mask=0
  g1[1] = (RADON_W & 0xFFFF) << 16;             // tensor_dim0 = 512
  g1[2] = (rows_remaining & 0xFFFF) << 16;      // tensor_dim1[15:0]
  g1[3] = (RADON_W << 16)                       // tile_dim0 = 512
        | ((rows_remaining >> 16) & 0xFFFF);    // tensor_dim1[31:16]
  g1[4] = BAND_ROWS;                            // tile_dim1 = 65, tile_dim2=0
  g1[5] = RADON_W;                              // tensor_dim0_stride = 512
  g1[6] = 0;                                    // stride0 hi / stride1 lo
  g1[7] = 0;                                    // stride1 hi
  tdm_v4i gz4; gz4[0] = 0; gz4[1] = 0; gz4[2] = 0; gz4[3] = 0;  // groups 2/3 unused
  tdm_v8i gz8;
  gz8[0] = 0; gz8[1] = 0; gz8[2] = 0; gz8[3] = 0;
  gz8[4] = 0; gz8[5] = 0; gz8[6] = 0; gz8[7] = 0;
  // 6-arg form (clang-23 / therock headers): (g0, g1, g2, g3, g4, cpol)
  __builtin_amdgcn_tensor_load_to_lds(g0, g1, gz4, gz4, gz8, 0);
}

__global__ __launch_bounds__(512, 1)
void radon_tdm_kernel(const float* __restrict__ x,
                      const int* __restrict__ theta,
                      float* __restrict__ out) {
  extern __shared__ float smem[];               // 2 band buffers
  const int blk = blockIdx.x;                   // n*180 + a
  const int n = blk / RADON_A;
  const int a = blk % RADON_A;
  const int w = threadIdx.x;                    // detector position

  const float ang = (float)theta[a] * 0.017453292519943295f;
  const float cs = cosf(ang);
  const float sn = sinf(ang);

  // pixel-space line: ix(h) = ixb + h*sn ; iy(h) = iyb + h*cs
  const float xw  = (2.0f * (float)w + 1.0f) * (1.0f / 512.0f) - 1.0f;
  const float ixb = 256.0f * cs * xw + sn * (0.5f - 256.0f) + 255.5f;
  const float iyb = -256.0f * sn * xw + cs * (0.5f - 256.0f) + 255.5f;

  const float* img = x + (size_t)n * RADON_H * RADON_W;
  const unsigned smem_off = (unsigned)(size_t)(void*)smem;  // LDS byte offset
  const bool isWave0 = (threadIdx.x < 32u);

  // Prologue: prefetch bands 0 and 1, wait for band 0.
  if (isWave0) {
    tdm_load_band(img + 0 * BAND_R * RADON_W, smem_off + 0 * BUF_BYTES,
                  RADON_H - 0 * BAND_R);
    if (NBANDS > 1)
      tdm_load_band(img + 1 * BAND_R * RADON_W, smem_off + 1 * BUF_BYTES,
                    RADON_H - 1 * BAND_R);
    __builtin_amdgcn_s_wait_tensorcnt(1);
  }
  __syncthreads();

  float acc = 0.0f;
  for (int k = 0; k < NBANDS; ++k) {
    const float* buf = smem + (k & 1) * BUF_FLOATS;
    const int b = k * BAND_R;

    // Analytic h-interval with floor(iy) in [own_lo, b+BAND_R-1].
    // Band 0 also owns floor(iy) == -1 (top corner zero-weighted below).
    const float lo = (k == 0) ? -1.0f : (float)b;
    const float hi = (float)(b + BAND_R);
    float h0f, h1f;
    if (cs > 1e-6f) {
      h0f = ceilf((lo - iyb) / cs);
      h1f = ceilf((hi - iyb) / cs);
    } else if (cs < -1e-6f) {
      h0f = floorf((hi - iyb) / cs) + 1.0f;
      h1f = floorf((lo - iyb) / cs) + 1.0f;
    } else {
      const bool in = (iyb >= lo) && (iyb < hi);
      h0f = 0.0f;
      h1f = in ? 512.0f : 0.0f;
    }
    const int h0 = (int)fminf(fmaxf(h0f, 0.0f), 512.0f);
    const int h1 = (int)fminf(fmaxf(h1f, 0.0f), 512.0f);

    for (int h = h0; h < h1; ++h) {
      // Recompute from base each step (no loop-carried FP drift vs reference).
      const float ix = fmaf((float)h, sn, ixb);
      const float iy = fmaf((float)h, cs, iyb);
      const float fx = floorf(ix);
      const float fy = floorf(iy);
      const float wx = ix - fx;
      const float wy = iy - fy;
      const int x0 = (int)fx;
      const int y0 = (int)fy - b;               // local band row, [-1, R-1]
      const float vx0 = (x0 >= 0 && x0 <= RADON_W - 1) ? 1.0f : 0.0f;
      const float vx1 = (x0 >= -1 && x0 <= RADON_W - 2) ? 1.0f : 0.0f;
      const float vy0 = (y0 >= 0) ? 1.0f : 0.0f;  // only false for floor==-1
      const int xc0 = min(max(x0, 0), RADON_W - 1);
      const int xc1 = min(max(x0 + 1, 0), RADON_W - 1);
      const int yc0 = max(y0, 0);
      const int y1 = y0 + 1;                    // in [0, R]; halo row is valid
      const float* r0 = buf + yc0 * RADON_W;
      const float* r1 = buf + y1 * RADON_W;
      const float omwx = 1.0f - wx;
      const float omwy = 1.0f - wy;
      acc = fmaf(r0[xc0], omwy * omwx * vy0 * vx0, acc);
      acc = fmaf(r0[xc1], omwy * wx * vy0 * vx1, acc);
      acc = fmaf(r1[xc0], wy * omwx * vx0, acc);   // bottom: zero-filled halo
      acc = fmaf(r1[xc1], wy * wx * vx1, acc);
    }

    __syncthreads();                            // all waves done reading buf
    if (isWave0) {
      const int nk = k + 2;
      if (nk < NBANDS)
        tdm_load_band(img + nk * BAND_R * RADON_W,
                      smem_off + (nk & 1) * BUF_BYTES,
                      RADON_H - nk * BAND_R);
      if (k + 1 < NBANDS) {
        if (nk < NBANDS) __builtin_amdgcn_s_wait_tensorcnt(1);  // band k+1 done
        else             __builtin_amdgcn_s_wait_tensorcnt(0);
      }
    }
    __syncthreads();                            // next band published
  }

  // out layout [N, C=1, W, A]
  out[((size_t)n * RADON_W + w) * RADON_A + a] = acc;
}

extern "C" void kernel_launch(void* const* d_in, const int* in_sizes, int n_in,
                              void* d_out, int out_size, void* d_ws, size_t ws_size,
                              hipStream_t stream) {
  (void)in_sizes; (void)n_in; (void)out_size; (void)d_ws; (void)ws_size;
  const float* x = (const float*)d_in[0];
  const int* theta = (const int*)d_in[1];
  float* out = (float*)d_out;
  dim3 grid(RADON_N * RADON_A);   // 360 workgroups: one per (batch, angle)
  dim3 block(RADON_W);            // 512 threads: one per detector position
  size_t shmem = 2 * BUF_BYTES;   // 260 KB double-buffered bands in LDS
  radon_tdm_kernel<<<grid, block, shmem, stream>>>(x, theta, out);
}